// myYOLO_33251636805723
// MI455X (gfx1250) — compile-verified
//
#include <hip/hip_runtime.h>
#include <stdint.h>

#define N_BOX      6400
#define NUM_CLS    20
#define GRID       80
#define STRIDEPX   32.0f
#define INPUT_SZ   2560.0f
#define CONF_TH    0.01f
#define EPS_NMS    1e-28f
#define MASK_WORDS 200          /* 6400 / 32  */
#define ROW_TILES  400          /* 6400 / 16  */
#define SORT_N     8192

typedef __attribute__((ext_vector_type(2))) float v2f;
typedef __attribute__((ext_vector_type(8))) float v8f;

__device__ __forceinline__ float sigmoidf(float x) { return 1.0f / (1.0f + __expf(-x)); }

// ---------------------------------------------------------------------------
// Kernel 1: per-box decode: conf*softmax max/argmax, box decode, offset boxes
// ---------------------------------------------------------------------------
__global__ void yolo_decode(const float* __restrict__ pred,
                            float* __restrict__ out_boxes,   // d_out[0 .. 25600)
                            float* __restrict__ boxo,        // class-offset boxes
                            float* __restrict__ area,
                            float* __restrict__ score,
                            int*   __restrict__ cls)
{
    int t = blockIdx.x * blockDim.x + threadIdx.x;
    if (t >= N_BOX) return;
    const float* row = pred + t * 25;

    float conf = sigmoidf(row[0]);

    float m = row[1]; int am = 0;
    #pragma unroll
    for (int c = 1; c < NUM_CLS; ++c) {
        float l = row[1 + c];
        if (l > m) { m = l; am = c; }          // first-max like jnp.argmax
    }
    float s = 0.0f;
    #pragma unroll
    for (int c = 0; c < NUM_CLS; ++c) s += __expf(row[1 + c] - m);
    float sc = conf / s;                        // max(softmax)*conf

    float gx = (float)(t / GRID), gy = (float)(t % GRID);
    float cx = (gx + sigmoidf(row[21])) * STRIDEPX;
    float cy = (gy + sigmoidf(row[22])) * STRIDEPX;
    float w  = __expf(row[23]);
    float h  = __expf(row[24]);

    float x1 = fminf(fmaxf((cx - 0.5f * w) / INPUT_SZ, 0.0f), 1.0f);
    float y1 = fminf(fmaxf((cy - 0.5f * h) / INPUT_SZ, 0.0f), 1.0f);
    float x2 = fminf(fmaxf((cx + 0.5f * w) / INPUT_SZ, 0.0f), 1.0f);
    float y2 = fminf(fmaxf((cy + 0.5f * h) / INPUT_SZ, 0.0f), 1.0f);

    out_boxes[4 * t + 0] = x1;
    out_boxes[4 * t + 1] = y1;
    out_boxes[4 * t + 2] = x2;
    out_boxes[4 * t + 3] = y2;

    float off = 2.0f * (float)am;               // class-offset trick
    float bx1 = x1 + off, by1 = y1 + off, bx2 = x2 + off, by2 = y2 + off;
    boxo[4 * t + 0] = bx1;
    boxo[4 * t + 1] = by1;
    boxo[4 * t + 2] = bx2;
    boxo[4 * t + 3] = by2;
    area[t]  = (bx2 - bx1) * (by2 - by1);
    score[t] = sc;
    cls[t]   = am;
}

// ---------------------------------------------------------------------------
// Kernel 2: single-workgroup bitonic argsort, descending score / ascending idx
// ---------------------------------------------------------------------------
__global__ void sort_scores(const float* __restrict__ score,
                            int* __restrict__ ord)
{
    __shared__ uint32_t skey[SORT_N];   // 32 KB
    __shared__ uint16_t sidx[SORT_N];   // 16 KB
    int tid = threadIdx.x;              // 1024 threads

    for (int r = tid; r < SORT_N; r += 1024) {
        uint32_t key; uint16_t idx;
        if (r < N_BOX) {
            uint32_t fb = __float_as_uint(score[r]);
            key = (fb & 0x80000000u) ? ~fb : (fb | 0x80000000u);  // order-preserving
            idx = (uint16_t)r;
        } else {
            key = 0u; idx = 0xFFFFu;    // sorts last
        }
        skey[r] = key; sidx[r] = idx;
    }
    __syncthreads();

    for (int k = 2; k <= SORT_N; k <<= 1) {
        for (int j = k >> 1; j > 0; j >>= 1) {
            for (int i = tid; i < SORT_N; i += 1024) {
                int ixj = i ^ j;
                if (ixj > i) {
                    uint32_t ka = skey[i], kb = skey[ixj];
                    uint16_t ia = sidx[i], ib = sidx[ixj];
                    bool gt_ab = (ka > kb) || (ka == kb && ia < ib); // a before b (desc)
                    bool gt_ba = (kb > ka) || (kb == ka && ib < ia);
                    bool up = (i & k) == 0;
                    if ((up && gt_ba) || (!up && gt_ab)) {
                        skey[i] = kb; skey[ixj] = ka;
                        sidx[i] = ib; sidx[ixj] = ia;
                    }
                }
            }
            __syncthreads();
        }
    }
    for (int r = tid; r < N_BOX; r += 1024) ord[r] = (int)sidx[r];
}

// ---------------------------------------------------------------------------
// Kernel 2b: gather boxes/areas/scores into sorted order
// ---------------------------------------------------------------------------
__global__ void gather_sorted(const int* __restrict__ ord,
                              const float* __restrict__ boxo,
                              const float* __restrict__ area,
                              const float* __restrict__ score,
                              float* __restrict__ sboxo,
                              float* __restrict__ sarea,
                              float* __restrict__ sscore)
{
    int r = blockIdx.x * blockDim.x + threadIdx.x;
    if (r >= N_BOX) return;
    int o = ord[r];
    sscore[r] = score[o];
    sarea[r]  = area[o];
    reinterpret_cast<float4*>(sboxo)[r] = reinterpret_cast<const float4*>(boxo)[o];
}

// ---------------------------------------------------------------------------
// Kernel 3: pairwise IoU suppression bitmap.
// One wave32 -> 16 rows x 32 cols tile -> one u32 mask word per row.
// areas_i + areas_j computed EXACTLY on the matrix pipe with
// v_wmma_f32_16x16x4_f32 (rank-2 outer-sum: products are x*1, exact).
// iou > 0.5  <=>  3*inter > (area_i + area_j).
// ---------------------------------------------------------------------------
__global__ void iou_mask(const float* __restrict__ sboxo,
                         const float* __restrict__ sarea,
                         uint32_t* __restrict__ mask)
{
    int lane = threadIdx.x & 31;
    int wave = (blockIdx.x * blockDim.x + threadIdx.x) >> 5;
    int rowTile = wave / MASK_WORDS;
    int wordIdx = wave - rowTile * MASK_WORDS;
    if (rowTile >= ROW_TILES) return;

    int i0 = rowTile * 16;
    int j0 = wordIdx * 32;

    if (j0 + 31 <= i0) {                 // tile entirely at/below diagonal
        if (lane < 16)
            mask[(size_t)(i0 + lane) * MASK_WORDS + wordIdx] = 0u;
        return;
    }

    const float4* sb4 = reinterpret_cast<const float4*>(sboxo);
    int half = lane >> 4;                // 0: lanes 0-15, 1: lanes 16-31
    int lm   = lane & 15;

    // A (16x4 f32): row M = [area_i, 1, 0, 0]. Lanes 0-15 hold K=0,1; 16-31 hold K=2,3 (=0).
    v2f a;
    a.x = half ? 0.0f : sarea[i0 + lm];
    a.y = half ? 0.0f : 1.0f;

    // 8 row boxes this lane is responsible for (M = v + 8*half)
    int rbase = i0 + half * 8;
    float4 rb[8];
    #pragma unroll
    for (int v = 0; v < 8; ++v) rb[v] = sb4[rbase + v];

    uint32_t bal[2][8];
    #pragma unroll
    for (int t = 0; t < 2; ++t) {
        int jbase = j0 + t * 16;
        // B (4x16 f32): col N = [1, area_j, 0, 0]^T
        v2f b;
        b.x = half ? 0.0f : 1.0f;
        b.y = half ? 0.0f : sarea[jbase + lm];
        v8f c = {};
        c = __builtin_amdgcn_wmma_f32_16x16x4_f32(
                /*neg_a=*/false, a, /*neg_b=*/false, b,
                /*c_mod=*/(short)0, c, /*reuse_a=*/false, /*reuse_b=*/false);

        float4 cb = sb4[jbase + lm];     // this lane's column box
        int jg = jbase + lm;
        #pragma unroll
        for (int v = 0; v < 8; ++v) {
            int ig = rbase + v;
            float xx1 = fmaxf(rb[v].x, cb.x);
            float yy1 = fmaxf(rb[v].y, cb.y);
            float xx2 = fminf(rb[v].z, cb.z);
            float yy2 = fminf(rb[v].w, cb.w);
            float iw  = fmaxf(EPS_NMS, xx2 - xx1);
            float ih  = fmaxf(EPS_NMS, yy2 - yy1);
            float inter = iw * ih;
            // c[v] = area_i + area_j for this lane's (M=v+8*half, N=lm) pair
            bool bit = (3.0f * inter > c[v]) && (jg > ig);
            bal[t][v] = (uint32_t)(__ballot(bit) & 0xFFFFFFFFull);
        }
    }

    if (lane < 16) {
        uint32_t wd = 0u;
        #pragma unroll
        for (int v = 0; v < 8; ++v) {
            uint32_t w_lo = (bal[0][v] & 0x0000FFFFu) | (bal[1][v] << 16);          // row v
            uint32_t w_hi = (bal[0][v] >> 16)         | (bal[1][v] & 0xFFFF0000u);  // row v+8
            wd = (lane == v)     ? w_lo : wd;
            wd = (lane == v + 8) ? w_hi : wd;
        }
        mask[(size_t)(i0 + lane) * MASK_WORDS + wordIdx] = wd;
    }
}

// ---------------------------------------------------------------------------
// Kernel 4: serial greedy NMS scan over the 200-word keep bitmap in LDS,
// then scatter fixed-shape outputs back to original box order.
// ---------------------------------------------------------------------------
__global__ void nms_scan(const uint32_t* __restrict__ mask,
                         const float* __restrict__ sscore,
                         const int* __restrict__ ord,
                         const int* __restrict__ cls,
                         float* __restrict__ out)
{
    __shared__ uint32_t keep[MASK_WORDS];
    int tid = threadIdx.x;               // 256 threads

    for (int w = tid; w < MASK_WORDS; w += 256) {
        uint32_t bits = 0u;
        for (int b = 0; b < 32; ++b)
            bits |= (sscore[w * 32 + b] >= CONF_TH ? 1u : 0u) << b;
        keep[w] = bits;                  // keep starts as the conf-valid mask
    }
    __syncthreads();

    for (int i = 0; i < N_BOX; ++i) {
        bool alive = (keep[i >> 5] >> (i & 31)) & 1u;   // uniform across block
        __syncthreads();
        if (alive) {
            const uint32_t* row = mask + (size_t)i * MASK_WORDS;
            for (int w = tid; w < MASK_WORDS; w += 256)
                keep[w] &= ~row[w];
        }
        if (i + 1 < N_BOX && tid < 13)   // warm next row (800 B) -> global_prefetch_b8
            __builtin_prefetch(mask + (size_t)(i + 1) * MASK_WORDS + tid * 16, 0, 0);
        __syncthreads();
    }

    for (int r = tid; r < N_BOX; r += 256) {
        bool k = (keep[r >> 5] >> (r & 31)) & 1u;
        int o = ord[r];
        out[25600 + o] = k ? sscore[r] : 0.0f;   // scores * keep
        out[32000 + o] = (float)cls[o];          // class_idx
        out[38400 + o] = k ? 1.0f : 0.0f;        // keep
    }
}

// ---------------------------------------------------------------------------
extern "C" void kernel_launch(void* const* d_in, const int* in_sizes, int n_in,
                              void* d_out, int out_size, void* d_ws, size_t ws_size,
                              hipStream_t stream)
{
    (void)in_sizes; (void)n_in; (void)out_size; (void)ws_size;

    const float* pred = (const float*)d_in[0];
    float* out = (float*)d_out;
    char* ws = (char*)d_ws;

    // workspace layout (16B aligned offsets), total ~5.5 MB
    float*    boxo   = (float*)(ws + 0);        // 6400*4 f32
    float*    area   = (float*)(ws + 102400);   // 6400 f32
    float*    score  = (float*)(ws + 128000);   // 6400 f32
    int*      cls    = (int*)  (ws + 153600);   // 6400 i32
    int*      ord    = (int*)  (ws + 179200);   // 6400 i32
    float*    sboxo  = (float*)(ws + 204800);   // 6400*4 f32 (sorted)
    float*    sarea  = (float*)(ws + 307200);   // 6400 f32   (sorted)
    float*    sscore = (float*)(ws + 332800);   // 6400 f32   (sorted)
    uint32_t* mask   = (uint32_t*)(ws + 358400);// 6400*200 u32 = 5.12 MB

    yolo_decode  <<<(N_BOX + 255) / 256, 256, 0, stream>>>(pred, out, boxo, area, score, cls);
    sort_scores  <<<1, 1024, 0, stream>>>(score, ord);
    gather_sorted<<<(N_BOX + 255) / 256, 256, 0, stream>>>(ord, boxo, area, score,
                                                           sboxo, sarea, sscore);
    iou_mask     <<<(ROW_TILES * MASK_WORDS * 32) / 256, 256, 0, stream>>>(sboxo, sarea, mask);
    nms_scan     <<<1, 256, 0, stream>>>(mask, sscore, ord, cls, out);
}